// SimpleHypergraphConv_27865747817132
// MI455X (gfx1250) — compile-verified
//
#include <hip/hip_runtime.h>

// SimpleHypergraphConv forward: out = sparse_coo(indices, values, [N,N]) @ x
//   indices: int64 [2, NNZ]  (row 0 = dst rows, row 1 = src cols)
//   values : f32   [NNZ]
//   x      : f32   [N, 32]
//   out    : f32   [N, 32]
//
// MI455X strategy: x (12.8MB) and out (12.8MB) are L2-resident (192MB L2), so
// the gather and the 51.2M f32 atomics resolve at L2; HBM sees only the 32MB
// edge stream. One lane per feature dim (D==32==wave32), 8 edges per wave for
// MLP: edge metadata via wave-uniform scalar loads (wide s_load clauses on the
// KMcnt path), 8 independent coalesced 128B gathers in flight, then 8
// fire-and-forget global_atomic_add_f32 (STOREcnt, no return).

#define FEAT_D          32
#define EDGES_PER_WAVE  8
#define WAVES_PER_BLOCK 8   // 256 threads = 8 wave32

__global__ __launch_bounds__(256)
void hgconv_zero_kernel(float4* __restrict__ out4, int n4) {
    int i = blockIdx.x * blockDim.x + threadIdx.x;
    if (i < n4) out4[i] = make_float4(0.f, 0.f, 0.f, 0.f);
}

__global__ __launch_bounds__(256)
void hgconv_scatter_kernel(const long long* __restrict__ rows,
                           const long long* __restrict__ cols,
                           const float*     __restrict__ vals,
                           const float*     __restrict__ x,
                           float*           __restrict__ out,
                           int nnz) {
    const int lane = threadIdx.x & 31;

    // Global wave id forced into an SGPR so per-edge metadata addresses are
    // provably wave-uniform -> scalar memory (s_load) path.
    int wave = blockIdx.x * WAVES_PER_BLOCK + (threadIdx.x >> 5);
    wave = __builtin_amdgcn_readfirstlane(wave);

    const int ebase = wave * EDGES_PER_WAVE;

    if (ebase + EDGES_PER_WAVE <= nnz) {
        // Fast path: all 8 edges valid. Consecutive e -> the compiler merges
        // metadata fetches into wide scalar-load clauses, and all 8 gathers
        // issue before any use (max memory-level parallelism).
        int   row[EDGES_PER_WAVE];
        int   col[EDGES_PER_WAVE];
        float v  [EDGES_PER_WAVE];
        float xv [EDGES_PER_WAVE];

#pragma unroll
        for (int i = 0; i < EDGES_PER_WAVE; ++i) {
            row[i] = (int)rows[ebase + i];
            col[i] = (int)cols[ebase + i];
            v[i]   = vals[ebase + i];
        }
#pragma unroll
        for (int i = 0; i < EDGES_PER_WAVE; ++i) {
            xv[i] = x[col[i] * FEAT_D + lane];       // coalesced 128B, L2 hit
        }
#pragma unroll
        for (int i = 0; i < EDGES_PER_WAVE; ++i) {
            atomicAdd(&out[row[i] * FEAT_D + lane], v[i] * xv[i]);
        }
    } else {
        // Tail path (not taken for NNZ=1.6M with this grid, kept for safety).
        for (int i = 0; i < EDGES_PER_WAVE; ++i) {
            const int e = ebase + i;
            if (e < nnz) {
                const int   r  = (int)rows[e];
                const int   c  = (int)cols[e];
                const float vv = vals[e];
                const float xx = x[c * FEAT_D + lane];
                atomicAdd(&out[r * FEAT_D + lane], vv * xx);
            }
        }
    }
}

extern "C" void kernel_launch(void* const* d_in, const int* in_sizes, int n_in,
                              void* d_out, int out_size, void* d_ws, size_t ws_size,
                              hipStream_t stream) {
    // Inputs per reference setup_inputs() order:
    //   d_in[0]: indices int64 [2, NNZ]   (in_sizes[0] == 2*NNZ int64 elems)
    //   d_in[1]: values  f32   [NNZ]
    //   d_in[2]: x       f32   [N, 32]
    const long long* indices = (const long long*)d_in[0];
    const int nnz            = in_sizes[0] / 2;
    const long long* rows    = indices;          // indices[0, :]
    const long long* cols    = indices + nnz;    // indices[1, :]
    const float* vals        = (const float*)d_in[1];
    const float* x           = (const float*)d_in[2];
    float* out               = (float*)d_out;

    // 1) Zero the output (harness poisons d_out; atomics need a clean base).
    {
        int n4     = out_size >> 2;              // out_size divisible by 4
        int blocks = (n4 + 255) / 256;
        hgconv_zero_kernel<<<blocks, 256, 0, stream>>>((float4*)out, n4);
    }

    // 2) Scatter-add all edges.
    {
        int waves  = (nnz + EDGES_PER_WAVE - 1) / EDGES_PER_WAVE;
        int blocks = (waves + WAVES_PER_BLOCK - 1) / WAVES_PER_BLOCK;
        hgconv_scatter_kernel<<<blocks, 256, 0, stream>>>(rows, cols, vals, x,
                                                          out, nnz);
    }
}